// Empanada2DInference_74113955660110
// MI455X (gfx1250) — compile-verified
//
#include <hip/hip_runtime.h>
#include <stdint.h>

#define HH 1024
#define WW 1024
#define NPIX (HH * WW)
#define THRESH 0.1f
#define KMAX 128
#define CAP 131072
#define NBLK 64
#define TILES_PER_WAVE 4  // 64 pixels per wave, A operand reused across 32 WMMAs

typedef unsigned int u32;
typedef unsigned long long u64;
typedef __attribute__((ext_vector_type(2))) float v2f;
typedef __attribute__((ext_vector_type(8))) float v8f;

// ---------------------------------------------------------------------------
// Workspace layout (bytes):
//   [0]        u32 survivor counter
//   [16]       u32 any-valid flag
//   [1024)     float A0[256]  (WMMA A operand, VGPR0 plane: K0 | K2)
//   [2048)     float A1[256]  (WMMA A operand, VGPR1 plane: K1 | K3)
//   [4096)     u64 list[CAP]          (1 MB survivor keys)
//   [1052672)  u64 cand[NBLK*KMAX]    (64 KB per-block top-128)
//   [2 MB)     float rmax[NPIX]       (4 MB separable NMS scratch)
// ---------------------------------------------------------------------------

// Pass 1: horizontal 7-tap max of thresholded heatmap; also zero the counter.
__global__ void nms_row_kernel(const float* __restrict__ hmp,
                               float* __restrict__ rmax,
                               u32* __restrict__ cnt) {
  int i = blockIdx.x * blockDim.x + threadIdx.x;
  if (i == 0) *cnt = 0u;
  if (i >= NPIX) return;
  int y = i >> 10, x = i & (WW - 1);
  float m = -3.0e38f;
#pragma unroll
  for (int dx = -3; dx <= 3; ++dx) {
    int xx = x + dx;
    if (xx < 0 || xx >= WW) continue;
    float v = hmp[(y << 10) + xx];
    v = (v > THRESH) ? v : -1.0f;
    m = fmaxf(m, v);
  }
  rmax[i] = m;
}

// Pass 2: vertical 7-tap max -> full 7x7 pooled; keep local maxima; append keys.
// Key = (score_bits << 32) | ~pixel_idx  => max() == (score desc, idx asc).
__global__ void nms_col_kernel(const float* __restrict__ hmp,
                               const float* __restrict__ rmax,
                               u64* __restrict__ list,
                               u32* __restrict__ cnt) {
  int i = blockIdx.x * blockDim.x + threadIdx.x;
  if (i >= NPIX) return;
  int y = i >> 10, x = i & (WW - 1);
  float m = -3.0e38f;
#pragma unroll
  for (int dy = -3; dy <= 3; ++dy) {
    int yy = y + dy;
    if (yy < 0 || yy >= HH) continue;
    m = fmaxf(m, rmax[(yy << 10) + x]);
  }
  float v = hmp[i];
  float t = (v > THRESH) ? v : -1.0f;
  if (t > 0.0f && t == m) {
    u32 p = atomicAdd(cnt, 1u);
    if (p < CAP) {
      u64 key = ((u64)__float_as_uint(t) << 32) | (u32)(~(u32)i);
      list[p] = key;
    }
  }
}

// Stage 1 of top-k: each block selects its slice's top-128 from LDS.
__global__ void topk_local_kernel(const u64* __restrict__ list,
                                  const u32* __restrict__ cnt,
                                  u64* __restrict__ cand) {
  __shared__ u64 skey[2048];  // CAP / NBLK max slice
  __shared__ u64 rk[256];
  __shared__ u32 rp[256];
  int tid = threadIdx.x;
  int b = blockIdx.x;
  u32 n = *cnt;
  if (n > CAP) n = CAP;
  u32 sliceLen = (n + NBLK - 1) / NBLK;
  u32 start = (u32)b * sliceLen;
  u32 end = start + sliceLen;
  if (start > n) start = n;
  if (end > n) end = n;
  u32 cl = end - start;
  for (u32 j = tid; j < cl; j += 256) skey[j] = list[start + j];
  __syncthreads();
  for (int k = 0; k < KMAX; ++k) {
    u64 bk = 0; u32 bp = 0;
    for (u32 j = tid; j < cl; j += 256) {
      u64 kk = skey[j];
      if (kk > bk) { bk = kk; bp = j; }
    }
    rk[tid] = bk; rp[tid] = bp;
    __syncthreads();
    for (int s = 128; s > 0; s >>= 1) {
      if (tid < s && rk[tid + s] > rk[tid]) { rk[tid] = rk[tid + s]; rp[tid] = rp[tid + s]; }
      __syncthreads();
    }
    if (tid == 0) {
      cand[b * KMAX + k] = rk[0];
      if (rk[0] != 0) skey[rp[0]] = 0;
    }
    __syncthreads();
  }
}

// Stage 2: global top-128 from 8192 candidates held in registers (8/thread),
// then emit ctr coords and the WMMA-A-layout center features.
__global__ void topk_final_kernel(const u64* __restrict__ cand,
                                  const u32* __restrict__ cnt,
                                  float* __restrict__ A0,
                                  float* __restrict__ A1,
                                  u32* __restrict__ anyv,
                                  float* __restrict__ ctr_out) {
  __shared__ u64 rk[1024];
  __shared__ u32 rp[1024];
  __shared__ u64 selKey[KMAX];
  __shared__ float sY[KMAX], sX[KMAX];
  __shared__ int sV[KMAX];
  int tid = threadIdx.x;
  u64 reg[8];
#pragma unroll
  for (int s = 0; s < 8; ++s) reg[s] = cand[tid * 8 + s];

  for (int k = 0; k < KMAX; ++k) {
    u64 bk = 0; u32 bp = 0;
#pragma unroll
    for (int s = 0; s < 8; ++s) {
      if (reg[s] > bk) { bk = reg[s]; bp = (u32)(tid * 8 + s); }
    }
    rk[tid] = bk; rp[tid] = bp;
    __syncthreads();
    for (int s = 512; s > 0; s >>= 1) {
      if (tid < s && rk[tid + s] > rk[tid]) { rk[tid] = rk[tid + s]; rp[tid] = rp[tid + s]; }
      __syncthreads();
    }
    u64 wk = rk[0]; u32 wp = rp[0];
    if (tid == 0) selKey[k] = wk;
    if (wk != 0 && (wp >> 3) == (u32)tid) {
      u32 slot = wp & 7u;
#pragma unroll
      for (int s = 0; s < 8; ++s)
        if ((u32)s == slot) reg[s] = 0;  // static indexing only -> stays in VGPRs
    }
    __syncthreads();
  }

  if (tid < KMAX) {
    u64 kk = selKey[tid];
    int valid = (kk != 0);
    u32 idx = ~((u32)kk);
    int y = valid ? (int)(idx >> 10) : 0;
    int x = valid ? (int)(idx & (WW - 1)) : 0;
    ctr_out[2 * tid] = (float)y;
    ctr_out[2 * tid + 1] = (float)x;
    sY[tid] = (float)y; sX[tid] = (float)x; sV[tid] = valid;
  }
  __syncthreads();
  // A-matrix (16x4 f32) layout: lanes 0-15 hold K0 (VGPR0) and K1 (VGPR1);
  // lanes 16-31 hold K2 (VGPR0) and K3 (VGPR1). Feature = [cy, cx, -(cy^2+cx^2), 0].
  if (tid < 256) {
    int lane = tid & 31;
    int c = (tid >> 5) * 16 + (lane & 15);
    float cy = sY[c], cx = sX[c];
    int v = sV[c];
    A0[tid] = (lane < 16) ? (v ? cy : 0.0f) : (v ? -(cy * cy + cx * cx) : -1.0e30f);
    A1[tid] = (lane < 16) ? (v ? cx : 0.0f) : 0.0f;
  }
  if (tid == 0) {
    u32 n = *cnt;
    *anyv = (n > 0) ? 1u : 0u;
  }
}

// Nearest-center assignment via V_WMMA_F32_16X16X4_F32.
// Per wave: 64 pixels (4 tiles of 16) x 128 centers. The A operand (all 128
// center features, 16 VGPRs) is loaded ONCE per wave and reused across all
// 4 tiles -> 32 WMMAs per wave, 4x less A traffic and setup overhead.
// Per tile: 8 WMMAs, register argmax over 8x8 scores/lane, one xor-16 merge.
__global__ void group_pixels_kernel(const float* __restrict__ off,
                                    const int* __restrict__ sem,
                                    const float* __restrict__ A0,
                                    const float* __restrict__ A1,
                                    const u32* __restrict__ anyv,
                                    float* __restrict__ out) {
  int gtid = blockIdx.x * blockDim.x + threadIdx.x;
  int wave = gtid >> 5;
  int lane = threadIdx.x & 31;
  int base0 = wave * (16 * TILES_PER_WAVE);  // exact grid: no partial waves
  int n = lane & 15;

  // Hoisted loop-invariant state: A operand (center features) + any-center flag.
  v2f a[8];
#pragma unroll
  for (int t = 0; t < 8; ++t) {
    a[t].x = A0[t * 32 + lane];
    a[t].y = A1[t * 32 + lane];
  }
  int anyC = (*anyv != 0u);

#pragma unroll
  for (int tile = 0; tile < TILES_PER_WAVE; ++tile) {
    int p = base0 + tile * 16 + n;
    float oy = off[p];
    float ox = off[NPIX + p];
    float py = (float)(p >> 10) + oy;
    float px = (float)(p & (WW - 1)) + ox;
    // B-matrix (4x16 f32): VGPR0 = rows K0|K2, VGPR1 = rows K1|K3 across halves.
    // Feature = [2*py, 2*px, 1, 0].
    v2f b;
    b.x = (lane < 16) ? (2.0f * py) : 1.0f;
    b.y = (lane < 16) ? (2.0f * px) : 0.0f;

    float best_s = -3.0e38f;
    int best_c = 0;
#pragma unroll
    for (int t = 0; t < 8; ++t) {
      v8f cacc = {};
      cacc = __builtin_amdgcn_wmma_f32_16x16x4_f32(
          /*neg_a=*/false, a[t], /*neg_b=*/false, b,
          /*c_mod=*/(short)0, cacc, /*reuse_a=*/false, /*reuse_b=*/false);
      int cb = t * 16 + ((lane >> 4) << 3);  // rows 0-7 (low half) / 8-15 (high half)
#pragma unroll
      for (int r = 0; r < 8; ++r) {
        float s = cacc[r];
        int cc = cb + r;
        if (s > best_s) { best_s = s; best_c = cc; }
      }
    }
    // Merge the two lane-halves (centers split 8/8 per column).
    float os = __shfl_xor(best_s, 16, 32);
    int oc = __shfl_xor(best_c, 16, 32);
    if (os > best_s || (os == best_s && oc < best_c)) { best_s = os; best_c = oc; }

    if (lane < 16) {
      int sv = sem[p];
      int thing = (sv == 1) || (sv == 2);
      out[p] = (thing && anyC) ? (float)(best_c + 1) : 0.0f;
    }
  }
}

extern "C" void kernel_launch(void* const* d_in, const int* in_sizes, int n_in,
                              void* d_out, int out_size, void* d_ws, size_t ws_size,
                              hipStream_t stream) {
  const int* sem = (const int*)d_in[0];
  const float* hmp = (const float*)d_in[1];
  const float* off = (const float*)d_in[2];
  float* out = (float*)d_out;
  float* ctr_out = out + NPIX;

  char* ws = (char*)d_ws;
  u32* cnt = (u32*)(ws + 0);
  u32* anyv = (u32*)(ws + 16);
  float* A0 = (float*)(ws + 1024);
  float* A1 = (float*)(ws + 2048);
  u64* list = (u64*)(ws + 4096);
  u64* cand = (u64*)(ws + 4096 + (size_t)CAP * 8);
  float* rmax = (float*)(ws + 2u * 1024u * 1024u);

  dim3 blk(256);
  dim3 grd(NPIX / 256);
  nms_row_kernel<<<grd, blk, 0, stream>>>(hmp, rmax, cnt);
  nms_col_kernel<<<grd, blk, 0, stream>>>(hmp, rmax, list, cnt);
  topk_local_kernel<<<dim3(NBLK), dim3(256), 0, stream>>>(list, cnt, cand);
  topk_final_kernel<<<dim3(1), dim3(1024), 0, stream>>>(cand, cnt, A0, A1, anyv, ctr_out);
  // NPIX / (16*TILES_PER_WAVE) waves * 32 lanes / 256 threads per block (exact).
  group_pixels_kernel<<<dim3(NPIX / (16 * TILES_PER_WAVE) * 32 / 256), blk, 0, stream>>>(
      off, sem, A0, A1, anyv, out);
}